// GNN_LRMC_88416196755744
// MI455X (gfx1250) — compile-verified
//
#include <hip/hip_runtime.h>

typedef __attribute__((ext_vector_type(2))) float v2f;
typedef __attribute__((ext_vector_type(8))) float v8f;

#define D_DIM 128

// h[m,o] = sum_k act(x[m,k]) * W[o,k] + b[o],  act = relu if RELU else identity.
// One 16x16 output tile per wave; 8 waves/block cover the 8 column tiles of D=128.
// WMMA f32 16x16x4: A/B VGPR v holds K = v + 2*(lane>>4); C VGPR v holds
// (M = v + 8*(lane>>4), N = lane&15).
template <bool RELU>
__global__ __launch_bounds__(256) void gnn_gemm_bias_wmma(
    const float* __restrict__ x, const float* __restrict__ Wm,
    const float* __restrict__ bias, float* __restrict__ h, int Nrows) {
  const int lane = threadIdx.x & 31;
  const int wave = threadIdx.x >> 5;   // col tile 0..7
  const int hi = lane >> 4;            // 0 or 1
  const int lo = lane & 15;
  const int m0 = blockIdx.x << 4;
  if (m0 >= Nrows) return;             // wave-uniform: EXEC stays all-ones for WMMA
  const int ocol = (wave << 4) + lo;

  const float* __restrict__ xrow = x + (size_t)(m0 + lo) * D_DIM;
  const float* __restrict__ wrow = Wm + (size_t)ocol * D_DIM;  // B[k][ocol] = W[ocol][k]

  const float bv = bias[ocol];
  v8f c;
#pragma unroll
  for (int v = 0; v < 8; ++v) c[v] = bv;

#pragma unroll 4
  for (int k = 0; k < D_DIM; k += 4) {
    const int ko = k + (hi << 1);
    v2f a, b;
    float a0 = xrow[ko];
    float a1 = xrow[ko + 1];
    if (RELU) {
      a0 = fmaxf(a0, 0.f);
      a1 = fmaxf(a1, 0.f);
    }
    a[0] = a0;
    a[1] = a1;
    b[0] = wrow[ko];
    b[1] = wrow[ko + 1];
    // 8 args: (neg_a, A, neg_b, B, c_mod, C, reuse_a, reuse_b)
    c = __builtin_amdgcn_wmma_f32_16x16x4_f32(false, a, false, b, (short)0, c,
                                              false, false);
  }

#pragma unroll
  for (int v = 0; v < 8; ++v)
    h[(size_t)(m0 + v + (hi << 3)) * D_DIM + ocol] = c[v];
}

// Wave-per-edge: lane l handles columns 4l..4l+3 (one b128 gather, 4 f32 atomics).
// Edge records staged through LDS so only 8 threads touch edge arrays per block.
// h table (51.2 MB) and agg (51.2 MB) are L2-resident on MI455X (192 MB L2), so
// the gathers and global_atomic_add_f32 resolve in L2, not HBM.
__global__ __launch_bounds__(256) void gnn_scatter(
    const int* __restrict__ src, const int* __restrict__ dst,
    const float* __restrict__ ew, const float* __restrict__ h,
    float* __restrict__ agg, int E) {
  __shared__ int s_s[8];
  __shared__ int s_d[8];
  __shared__ float s_w[8];

  const int t = threadIdx.x;
  const int ebase = blockIdx.x * 8;
  if (t < 8) {
    const int e = ebase + t;
    if (e < E) {
      s_s[t] = src[e];
      s_d[t] = dst[e];
      s_w[t] = ew[e];
    }
  }
  __syncthreads();

  const int w = t >> 5;
  if (ebase + w >= E) return;
  const int lane = t & 31;

  const int sIdx = s_s[w];
  const int dIdx = s_d[w];
  const float wgt = s_w[w];

  const float4 m = *((const float4*)(h + (size_t)sIdx * D_DIM) + lane);
  float* ap = agg + (size_t)dIdx * D_DIM + (lane << 2);

  const float r0 = wgt * m.x, r1 = wgt * m.y, r2 = wgt * m.z, r3 = wgt * m.w;
  // Pin the hardware L2 f32 atomic path (no CAS loop), device scope.
  asm volatile(
      "global_atomic_add_f32 %0, %1, off offset:0 scope:SCOPE_DEV\n\t"
      "global_atomic_add_f32 %0, %2, off offset:4 scope:SCOPE_DEV\n\t"
      "global_atomic_add_f32 %0, %3, off offset:8 scope:SCOPE_DEV\n\t"
      "global_atomic_add_f32 %0, %4, off offset:12 scope:SCOPE_DEV"
      :
      : "v"(ap), "v"(r0), "v"(r1), "v"(r2), "v"(r3)
      : "memory");
}

__global__ __launch_bounds__(256) void gnn_relu(const float4* __restrict__ agg,
                                                float4* __restrict__ x, int n4) {
  int i = blockIdx.x * blockDim.x + threadIdx.x;
  if (i < n4) {
    float4 v = agg[i];
    v.x = fmaxf(v.x, 0.f);
    v.y = fmaxf(v.y, 0.f);
    v.z = fmaxf(v.z, 0.f);
    v.w = fmaxf(v.w, 0.f);
    x[i] = v;
  }
}

extern "C" void kernel_launch(void* const* d_in, const int* in_sizes, int n_in,
                              void* d_out, int out_size, void* d_ws, size_t ws_size,
                              hipStream_t stream) {
  const int* edge_index = (const int*)d_in[0];   // [2, E] flat
  const float* edge_w   = (const float*)d_in[1]; // [E]
  const float* row_emb  = (const float*)d_in[2]; // [N_ROWS, 128]
  const float* col_emb  = (const float*)d_in[3]; // [N_COLS, 128]
  const float* Wall     = (const float*)d_in[4]; // [L, 128, 128]
  const float* ball     = (const float*)d_in[5]; // [L, 128]

  const int E     = in_sizes[1];
  const int N_R   = in_sizes[2] / D_DIM;
  const int N_C   = in_sizes[3] / D_DIM;
  const int N     = N_R + N_C;
  const int L     = in_sizes[5] / D_DIM;
  const int nelem = N * D_DIM;
  const int n4    = nelem / 4;

  float* xbuf = (float*)d_out;                 // layer-0 input + final output
  float* h    = (float*)d_ws;                  // [N, 128]
  float* agg  = (float*)d_ws + (size_t)nelem;  // [N, 128]

  const int* src = edge_index;
  const int* dst = edge_index + E;

  // x0 = concat(row_embed, col_embed)
  hipMemcpyAsync(xbuf, row_emb, (size_t)N_R * D_DIM * sizeof(float),
                 hipMemcpyDeviceToDevice, stream);
  hipMemcpyAsync(xbuf + (size_t)N_R * D_DIM, col_emb,
                 (size_t)N_C * D_DIM * sizeof(float),
                 hipMemcpyDeviceToDevice, stream);

  const int gemm_blocks    = (N + 15) / 16;
  const int vec_blocks     = (n4 + 255) / 256;
  const int scatter_blocks = (E + 7) / 8;

  for (int i = 0; i < L; ++i) {
    const float* Wi = Wall + (size_t)i * D_DIM * D_DIM;
    const float* bi = ball + (size_t)i * D_DIM;

    if (i == 0) {
      gnn_gemm_bias_wmma<false><<<gemm_blocks, 256, 0, stream>>>(xbuf, Wi, bi, h, N);
    } else {
      // ReLU of previous layer fused into the A-fragment load
      gnn_gemm_bias_wmma<true><<<gemm_blocks, 256, 0, stream>>>(agg, Wi, bi, h, N);
    }
    hipMemsetAsync(agg, 0, (size_t)nelem * sizeof(float), stream);
    gnn_scatter<<<scatter_blocks, 256, 0, stream>>>(src, dst, edge_w, h, agg, E);
  }

  // final activation -> output
  gnn_relu<<<vec_blocks, 256, 0, stream>>>((const float4*)agg, (float4*)xbuf, n4);
}